// Canny_82789789598318
// MI455X (gfx1250) — compile-verified
//
#include <hip/hip_runtime.h>
#include <hip/hip_bf16.h>
#include <stdint.h>
#include <math.h>

#define H_IMG 2048
#define W_IMG 2048
#define WORDS (W_IMG / 32)       // 64 bit-packed words per row
#define T_LOW_F 100.0f
#define T_HIGH_F 200.0f

typedef __attribute__((ext_vector_type(4))) unsigned int u32x4;
typedef __attribute__((ext_vector_type(8))) int          i32x8;
typedef __attribute__((ext_vector_type(4))) int          i32x4;
typedef __attribute__((ext_vector_type(4))) float        f32x4;

__device__ __forceinline__ int clampi(int v, int lo, int hi) {
  return v < lo ? lo : (v > hi ? hi : v);
}

// ---------------------------------------------------------------------------
// Tensor Data Mover helpers: build a 2D D# (data_size = 4B elements) and issue
// the DMA. Issue from a single lane; TDM ops ignore EXEC, so one call = one
// DMA for the workgroup. Completion tracked on TENSORcnt.
// ---------------------------------------------------------------------------
__device__ __forceinline__ void tdm_desc_2d(u32x4& g0, i32x8& g1,
                                            const void* gptr, unsigned ldsAddr,
                                            int tileW, int tileH,
                                            int strideElems,
                                            int tensorW, int tensorH) {
  unsigned long long ga = (unsigned long long)(uintptr_t)gptr;
  g0.x = 1u;                                        // count=1 (valid), user mode
  g0.y = ldsAddr;                                   // lds_addr (bytes)
  g0.z = (unsigned)(ga & 0xFFFFFFFFull);            // global_addr[31:0]
  g0.w = (unsigned)((ga >> 32) & 0x01FFFFFFull)     // global_addr[56:32]
         | (2u << 30);                              // type = 2 ("image")
  g1[0] = (2 << 16);                                // data_size=4B, wg_mask=0
  g1[1] = (int)((tensorW & 0xFFFF) << 16);          // tensor_dim0 lo (bits 63:48)
  g1[2] = (int)(((tensorW >> 16) & 0xFFFF) | ((tensorH & 0xFFFF) << 16));
  g1[3] = (int)(((tensorH >> 16) & 0xFFFF) | ((tileW & 0xFFFF) << 16));
  g1[4] = (int)(tileH & 0xFFFF);                    // tile_dim1, tile_dim2=0
  g1[5] = strideElems;                              // tensor_dim0_stride[31:0]
  g1[6] = 0;
  g1[7] = 0;
}

__device__ __forceinline__ void tdm_load_2d(const void* gptr, unsigned ldsAddr,
                                            int tileW, int tileH, int strideElems,
                                            int tensorW, int tensorH) {
  u32x4 g0; i32x8 g1;
  tdm_desc_2d(g0, g1, gptr, ldsAddr, tileW, tileH, strideElems, tensorW, tensorH);
  i32x4 zz4 = {0, 0, 0, 0};
  i32x8 zz8 = {0, 0, 0, 0, 0, 0, 0, 0};
  __builtin_amdgcn_tensor_load_to_lds(g0, g1, zz4, zz4, zz8, /*cpol=*/0);
}

__device__ __forceinline__ void tdm_store_2d(void* gptr, unsigned ldsAddr,
                                             int tileW, int tileH, int strideElems,
                                             int tensorW, int tensorH) {
  u32x4 g0; i32x8 g1;
  tdm_desc_2d(g0, g1, gptr, ldsAddr, tileW, tileH, strideElems, tensorW, tensorH);
  i32x4 zz4 = {0, 0, 0, 0};
  i32x8 zz8 = {0, 0, 0, 0, 0, 0, 0, 0};
  __builtin_amdgcn_tensor_store_from_lds(g0, g1, zz4, zz4, zz8, /*cpol=*/0);
}

// ---------------------------------------------------------------------------
// Kernel 1: TDM tile load -> quantize -> Sobel -> NMS -> double threshold.
// Each block owns a 32x32 tile; the TDM DMAs a 36x36 f32 halo region into LDS
// (mag needs halo 1, Sobel another 1). Tile origin is clamped inside the
// image; replicate-border is realized by clamped LDS indexing.
// ---------------------------------------------------------------------------
__global__ __launch_bounds__(256) void canny_grad_nms(
    const float* __restrict__ img,
    unsigned* __restrict__ strongBits,   // also the initial `cur`
    unsigned* __restrict__ weakBits) {
  __shared__ float simg[36 * 36];
  __shared__ float smag[34][34];

  const int bx = blockIdx.x, by = blockIdx.y;
  const int tid = threadIdx.x;
  const int tileX = bx * 32, tileY = by * 32;
  const int gx0 = clampi(tileX - 2, 0, W_IMG - 36);
  const int gy0 = clampi(tileY - 2, 0, H_IMG - 36);

  if (tid == 0) {
    tdm_load_2d(img + (size_t)gy0 * W_IMG + gx0,
                (unsigned)(uintptr_t)(&simg[0]),
                /*tileW=*/36, /*tileH=*/36, /*stride=*/W_IMG,
                /*tensorW=*/W_IMG, /*tensorH=*/H_IMG);
    __builtin_amdgcn_s_wait_tensorcnt(0);
  }
  __syncthreads();

  // Quantize in place: clip(floor(x*255), 0, 255)
  for (int i = tid; i < 36 * 36; i += 256) {
    simg[i] = fminf(fmaxf(floorf(simg[i] * 255.0f), 0.0f), 255.0f);
  }
  __syncthreads();

#define SAMP(dy, dx)                                              \
  simg[(clampi(gy + (dy), 0, H_IMG - 1) - gy0) * 36 +             \
       (clampi(gx + (dx), 0, W_IMG - 1) - gx0)]

  // L1 gradient magnitude over the 34x34 halo region (tile +/- 1).
  for (int m = tid; m < 34 * 34; m += 256) {
    int my = m / 34, mx = m % 34;
    int gy = tileY + my - 1, gx = tileX + mx - 1;
    float a = SAMP(-1, -1), b = SAMP(-1, 0), c = SAMP(-1, 1);
    float d = SAMP(0, -1),                   e = SAMP(0, 1);
    float f = SAMP(1, -1),  g = SAMP(1, 0),  h = SAMP(1, 1);
    float gxs = (c + 2.0f * e + h) - (a + 2.0f * d + f);
    float gys = (f + 2.0f * g + h) - (a + 2.0f * b + c);
    smag[my][mx] = fabsf(gxs) + fabsf(gys);
  }
  __syncthreads();

  // NMS + double threshold; one full 32-pixel row per wave -> ballot packs it.
  const int lane = tid & 31;
  const int rowgrp = tid >> 5;                       // 8 waves
  for (int i = 0; i < 4; ++i) {
    int ty = rowgrp + i * 8;
    int tx = lane;
    int gy = tileY + ty, gx = tileX + tx;
    int my = ty + 1, mx = tx + 1;

    float a = SAMP(-1, -1), b = SAMP(-1, 0), c = SAMP(-1, 1);
    float d = SAMP(0, -1),                   e = SAMP(0, 1);
    float f = SAMP(1, -1),  g = SAMP(1, 0),  h = SAMP(1, 1);
    float gxs = (c + 2.0f * e + h) - (a + 2.0f * d + f);
    float gys = (f + 2.0f * g + h) - (a + 2.0f * b + c);
    float mag = smag[my][mx];

    float ang = atan2f(gys, gxs) * 57.29577951308232f;
    ang = fmodf(ang, 180.0f);
    if (ang < 0.0f) ang += 180.0f;

    float n1, n2;
    if (ang < 22.5f || ang >= 157.5f) { n1 = smag[my][mx + 1];     n2 = smag[my][mx - 1]; }
    else if (ang < 67.5f)             { n1 = smag[my - 1][mx + 1]; n2 = smag[my + 1][mx - 1]; }
    else if (ang < 112.5f)            { n1 = smag[my - 1][mx];     n2 = smag[my + 1][mx]; }
    else                              { n1 = smag[my - 1][mx - 1]; n2 = smag[my + 1][mx + 1]; }

    bool interior = (gy > 0) && (gy < H_IMG - 1) && (gx > 0) && (gx < W_IMG - 1);
    bool keep = interior && (mag >= n1) && (mag > n2);
    bool strong = keep && (mag > T_HIGH_F);
    bool weak   = keep && (mag > T_LOW_F);

    unsigned sb = (unsigned)(__ballot(strong) & 0xFFFFFFFFull);
    unsigned wb = (unsigned)(__ballot(weak)   & 0xFFFFFFFFull);
    if (lane == 0) {
      strongBits[gy * WORDS + bx] = sb;
      weakBits[gy * WORDS + bx]   = wb;
    }
  }
#undef SAMP
}

// ---------------------------------------------------------------------------
// Kernel 2: one hysteresis pass. Bit-parallel 8-connected dilation:
// new = cur | (weak & dilate8(cur)). Each block: 16 rows x full width.
// Interior rows + weak rows arrive via TDM (contiguous 4KB tiles); halo rows
// (which may fall off the image) use plain loads. Write-back also via TDM.
// 24 local iterations per pass; stripe halos exchanged between global passes.
// Monotone and fixed-trip -> deterministic under graph capture.
// ---------------------------------------------------------------------------
#define PROWS 16
__global__ __launch_bounds__(256) void hyst_pass(
    const unsigned* __restrict__ src, unsigned* __restrict__ dst,
    const unsigned* __restrict__ weakBits) {
  __shared__ unsigned scur[PROWS + 2][WORDS];
  __shared__ unsigned swk[PROWS][WORDS];

  const int t = threadIdx.x;
  const int wc = t & (WORDS - 1);
  const int r4 = t >> 6;                      // 0..3
  const int rowBase = blockIdx.x * PROWS;

  // Halo rows (row -1 and row PROWS): may be outside the image -> plain loads.
  if (t < WORDS) {
    int gr = rowBase - 1;
    scur[0][t] = (gr >= 0) ? src[(size_t)gr * WORDS + t] : 0u;
  } else if (t < 2 * WORDS) {
    int w2 = t - WORDS;
    int gr = rowBase + PROWS;
    scur[PROWS + 1][w2] = (gr < H_IMG) ? src[(size_t)gr * WORDS + w2] : 0u;
  }
  // Interior rows + weak stripe via TDM (always fully in range).
  if (t == 0) {
    tdm_load_2d(src + (size_t)rowBase * WORDS,
                (unsigned)(uintptr_t)(&scur[1][0]),
                WORDS, PROWS, WORDS, WORDS, H_IMG);
    tdm_load_2d(weakBits + (size_t)rowBase * WORDS,
                (unsigned)(uintptr_t)(&swk[0][0]),
                WORDS, PROWS, WORDS, WORDS, H_IMG);
    __builtin_amdgcn_s_wait_tensorcnt(0);
  }
  __syncthreads();

  for (int it = 0; it < 24; ++it) {
    unsigned nv[4];
    for (int k = 0; k < 4; ++k) {
      int r = r4 + k * 4 + 1;
      unsigned lu = wc ? scur[r - 1][wc - 1] : 0u;
      unsigned cu = scur[r - 1][wc];
      unsigned ru = (wc < WORDS - 1) ? scur[r - 1][wc + 1] : 0u;
      unsigned lm = wc ? scur[r][wc - 1] : 0u;
      unsigned cm = scur[r][wc];
      unsigned rm = (wc < WORDS - 1) ? scur[r][wc + 1] : 0u;
      unsigned ld = wc ? scur[r + 1][wc - 1] : 0u;
      unsigned cd = scur[r + 1][wc];
      unsigned rd = (wc < WORDS - 1) ? scur[r + 1][wc + 1] : 0u;
      unsigned du = cu | (cu << 1) | (cu >> 1) | (lu >> 31) | (ru << 31);
      unsigned dm = cm | (cm << 1) | (cm >> 1) | (lm >> 31) | (rm << 31);
      unsigned dd = cd | (cd << 1) | (cd >> 1) | (ld >> 31) | (rd << 31);
      nv[k] = cm | (swk[r - 1][wc] & (du | dm | dd));
    }
    __syncthreads();
    for (int k = 0; k < 4; ++k) scur[r4 + k * 4 + 1][wc] = nv[k];
    __syncthreads();
  }

  // Write back the interior stripe with a single TDM store.
  if (t == 0) {
    tdm_store_2d(dst + (size_t)rowBase * WORDS,
                 (unsigned)(uintptr_t)(&scur[1][0]),
                 WORDS, PROWS, WORDS, WORDS, H_IMG);
    __builtin_amdgcn_s_wait_tensorcnt(0);   // keep LDS alive until DMA done
  }
}

// ---------------------------------------------------------------------------
// Kernel 3: bits -> 3xHxW float32 (the 48 MB that dominates runtime).
// 4 pixels/thread, float4 non-temporal stores (streaming write-once data:
// do not displace the L2-resident working set).
// ---------------------------------------------------------------------------
__global__ __launch_bounds__(256) void expand_bits(
    const unsigned* __restrict__ bits, float* __restrict__ out) {
  const int HW = H_IMG * W_IMG;
  int q = blockIdx.x * 256 + threadIdx.x;      // quad index over HW/4
  if (q < HW / 4) {
    int base = q * 4;
    unsigned w = bits[base >> 5];
    unsigned nib = (w >> (base & 31)) & 0xFu;
    f32x4 v;
    v.x = (nib & 1u) ? 1.0f : 0.0f;
    v.y = (nib & 2u) ? 1.0f : 0.0f;
    v.z = (nib & 4u) ? 1.0f : 0.0f;
    v.w = (nib & 8u) ? 1.0f : 0.0f;
    f32x4* o = (f32x4*)out + q;
    __builtin_nontemporal_store(v, o);
    __builtin_nontemporal_store(v, o + HW / 4);
    __builtin_nontemporal_store(v, o + 2 * (HW / 4));
  }
}

extern "C" void kernel_launch(void* const* d_in, const int* in_sizes, int n_in,
                              void* d_out, int out_size, void* d_ws, size_t ws_size,
                              hipStream_t stream) {
  const float* img = (const float*)d_in[0];
  float* out = (float*)d_out;

  unsigned char* ws = (unsigned char*)d_ws;
  const size_t MASK_BYTES = (size_t)H_IMG * WORDS * sizeof(unsigned);  // 512 KB
  unsigned* weakB = (unsigned*)(ws);
  unsigned* bufA  = (unsigned*)(ws + MASK_BYTES);
  unsigned* bufB  = (unsigned*)(ws + 2 * MASK_BYTES);

  canny_grad_nms<<<dim3(W_IMG / 32, H_IMG / 32), 256, 0, stream>>>(img, bufA, weakB);

  unsigned* s = bufA;
  unsigned* d = bufB;
  for (int p = 0; p < 10; ++p) {           // even count: result ends in bufA
    hyst_pass<<<H_IMG / PROWS, 256, 0, stream>>>(s, d, weakB);
    unsigned* tmp = s; s = d; d = tmp;
  }

  expand_bits<<<(H_IMG * W_IMG / 4 + 255) / 256, 256, 0, stream>>>(s, out);
}